// DictionaryLearning_12824772346354
// MI455X (gfx1250) — compile-verified
//
#include <hip/hip_runtime.h>
#include <hip/hip_bf16.h>

typedef __attribute__((ext_vector_type(2))) float v2f;
typedef __attribute__((ext_vector_type(8))) float v8f;

#define NUM_EMB 512
#define EMB_DIM 64
#define BCOLS   32768
#define SPARS   5
#define ZTOTAL  2097152   // 32*64*32*32
#define XT_STRIDE 68      // padded LDS stride (even -> 8B-aligned b64; %64==4 -> bank spread)

// Map (m, b) of ze_flat (64 x 32768, a plain reshape of z_perm) back to the
// original z_e (32,64,32,32) flat index. The output transpose collapses to the
// SAME formula, so the z_out store reuses this address.
__device__ __forceinline__ int zaddr(int m, int b) {
    int c = b & 63;
    int w = (b >> 6) & 31;
    int t = m * 16 + (b >> 11);
    int h = t & 31;
    int n = t >> 5;
    return (n << 16) | (c << 10) | (h << 5) | w;
}

// ---------- kernel 1: zero [loss | coeff] region of d_out + loss accumulator ----------
// Region starts at d_out+ZTOTAL which is 16B aligned -> b128 stores on the
// biggest single traffic stream (64 MB of the ~80 MB total HBM writes).
__global__ void dl_zero_kernel(float* __restrict__ region, float* __restrict__ acc) {
    size_t i = (size_t)blockIdx.x * blockDim.x + threadIdx.x;
    size_t stride = (size_t)gridDim.x * blockDim.x;
    size_t n4 = ((size_t)NUM_EMB * BCOLS + 1) / 4;   // 4,194,304 float4s
    float4* p = (float4*)region;
    float4 z = make_float4(0.f, 0.f, 0.f, 0.f);
    for (size_t j = i; j < n4; j += stride) p[j] = z;
    if (i == 0) {
        region[(size_t)NUM_EMB * BCOLS] = 0.f;       // tail element
        *acc = 0.f;
    }
}

// ---------- kernel 1b: Dt = D^T (512 atoms x 64 dims), one-shot 128 KB, L2-resident ----
// Gives k-adjacent b64 WMMA B-operand loads and coalesced reconstruction reads.
__global__ void dl_dt_kernel(const float* __restrict__ D, float* __restrict__ Dt) {
    int tid = blockIdx.x * blockDim.x + threadIdx.x;   // 32768 threads, 1 element each
    int k = tid & 63, n = tid >> 6;
    Dt[tid] = D[k * NUM_EMB + n];                      // coalesced write, strided read
}

// ---------- kernel 2: G = D^T D (512x512) via V_WMMA_F32_16X16X4_F32 ----------
// 1024 16x16 tiles, one wave each; K = 64 -> 16 WMMA steps. G stays L2-resident.
__global__ void __launch_bounds__(256)
dl_gram_kernel(const float* __restrict__ D, float* __restrict__ G) {
    int lane = threadIdx.x & 31;
    int wid  = blockIdx.x * (blockDim.x >> 5) + (threadIdx.x >> 5);
    int ti = wid >> 5, tj = wid & 31;
    int i0 = ti * 16, j0 = tj * 16;
    int mrow = lane & 15;
    int ksel = (lane >> 4) << 1;     // lanes 0-15: K=k0,k0+1 ; lanes 16-31: K=k0+2,k0+3
    v8f c = {};
    #pragma unroll
    for (int kc = 0; kc < 16; ++kc) {
        int k0 = kc * 4 + ksel;
        v2f a, b;
        a.x = D[(k0    ) * NUM_EMB + i0 + mrow];   // A = D^T tile (16 x 4)
        a.y = D[(k0 + 1) * NUM_EMB + i0 + mrow];
        b.x = D[(k0    ) * NUM_EMB + j0 + mrow];   // B = D tile (4 x 16)
        b.y = D[(k0 + 1) * NUM_EMB + j0 + mrow];
        c = __builtin_amdgcn_wmma_f32_16x16x4_f32(false, a, false, b, (short)0, c, false, false);
    }
    int Mb = (lane >> 4) << 3;       // C layout: lanes 0-15 M=j, lanes 16-31 M=j+8
    #pragma unroll
    for (int j = 0; j < 8; ++j)
        G[(i0 + j + Mb) * NUM_EMB + j0 + mrow] = c[j];
}

// ---------- kernel 3: fused h_bar GEMM (WMMA -> LDS) + batch OMP ----------
// One 256-thread block (8 waves) handles 32 columns. h_bar tile 32x512 f32 = 64 KB LDS,
// never spilled to HBM (saves 128 MB of round-trip traffic vs. unfused). The X tile is
// staged (transposed, padded) through the same LDS buffer before hbar overwrites it.
__global__ void __launch_bounds__(256)
dl_omp_kernel(const float* __restrict__ ze, const float* __restrict__ Dt,
              const float* __restrict__ G, float* __restrict__ zout,
              float* __restrict__ coeff, float* __restrict__ acc) {
    __shared__ float hbar[32 * NUM_EMB];   // 64 KB; first 32*68 floats double as X tile
    int lane = threadIdx.x & 31;
    int widx = threadIdx.x >> 5;
    int bblk = blockIdx.x * 32;

    // ---- stage 0: cooperative gather of X tile into LDS, layout xt[col][k] ----
    {
        float* xt = hbar;                  // overlay: consumed before hbar is written
        #pragma unroll
        for (int u = 0; u < 8; ++u) {
            int e = threadIdx.x + 256 * u; // 2048 = 64*32 elements, each gathered once
            int m = e & 63, lc = e >> 6;   // k fast -> conflict-light LDS stores
            xt[lc * XT_STRIDE + m] = ze[zaddr(m, bblk + lc)];
        }
    }
    __syncthreads();

    // ---- stage 1: h_bar[b, n] = sum_m X[m,b] * D[m,n] via WMMA f32 16x16x4 ----
    {
        int r    = widx & 1;               // which 16-row (column) subtile
        int grp  = widx >> 1;              // 4 groups x 8 atom-tiles = 32 atom-tiles
        int mrow = lane & 15;
        int ksel = (lane >> 4) << 1;
        const float* xt = hbar;
        v2f areg[16];                       // A (16 cols x 64 K), one ds_load_b64 per chunk
        #pragma unroll
        for (int kc = 0; kc < 16; ++kc) {
            int k0 = kc * 4 + ksel;
            areg[kc] = *(const v2f*)(xt + (r * 16 + mrow) * XT_STRIDE + k0);
        }
        __syncthreads();                    // all A regs read before hbar stores begin
        #pragma unroll
        for (int t = 0; t < 8; ++t) {
            int n0 = (grp * 8 + t) * 16;
            v8f cacc = {};
            #pragma unroll
            for (int kc = 0; kc < 16; ++kc) {
                int k0 = kc * 4 + ksel;
                // Dt row = one atom's 64 dims: k0,k0+1 adjacent -> single b64 load
                v2f bb = *(const v2f*)(Dt + (n0 + mrow) * EMB_DIM + k0);
                cacc = __builtin_amdgcn_wmma_f32_16x16x4_f32(false, areg[kc], false, bb,
                                                             (short)0, cacc, false, false);
            }
            int Mb = (lane >> 4) << 3;
            #pragma unroll
            for (int j = 0; j < 8; ++j)
                hbar[(r * 16 + j + Mb) * NUM_EMB + n0 + mrow] = cacc[j];
        }
    }
    __syncthreads();

    float lacc = 0.f;
    // ---- stage 2: OMP, 4 columns per wave, lanes cooperate across the 512 atoms ----
    for (int q = 0; q < 4; ++q) {
        int lc = widx * 4 + q;
        int b  = bblk + lc;
        const float* hb = &hbar[lc * NUM_EMB];
        int   I[SPARS];
        float xs[SPARS];
        float hsel[SPARS];
        float Lm[SPARS][SPARS];
        #pragma unroll
        for (int k = 0; k < SPARS; ++k) {
            // argmax_n |h_bar[n] - sum_{j<k} xs[j]*G[I[j],n]| over unselected n,
            // 4 atoms per lane via b128 loads of hbar (LDS) and G rows (L2).
            float bestV = -1.f; int bestI = 0;
            #pragma unroll
            for (int i = 0; i < NUM_EMB / 128; ++i) {
                int n0 = i * 128 + lane * 4;
                float4 hv = *(const float4*)(hb + n0);
                float v0 = hv.x, v1 = hv.y, v2 = hv.z, v3 = hv.w;
                #pragma unroll
                for (int j = 0; j < SPARS; ++j) {
                    if (j < k) {
                        float4 gv = *(const float4*)(G + I[j] * NUM_EMB + n0);
                        v0 -= xs[j] * gv.x; v1 -= xs[j] * gv.y;
                        v2 -= xs[j] * gv.z; v3 -= xs[j] * gv.w;
                    }
                }
                float av[4] = { fabsf(v0), fabsf(v1), fabsf(v2), fabsf(v3) };
                #pragma unroll
                for (int e = 0; e < 4; ++e) {
                    int n = n0 + e;
                    bool msk = false;
                    #pragma unroll
                    for (int j = 0; j < SPARS; ++j)
                        if (j < k) msk |= (n == I[j]);
                    float a = msk ? -1.f : av[e];
                    if (a > bestV) { bestV = a; bestI = n; }   // keep-first within lane
                }
            }
            #pragma unroll
            for (int off = 16; off > 0; off >>= 1) {          // wave32 reduce
                float ov = __shfl_xor(bestV, off);
                int   oi = __shfl_xor(bestI, off);
                if (ov > bestV || (ov == bestV && oi < bestI)) { bestV = ov; bestI = oi; }
            }
            I[k]    = bestI;
            hsel[k] = hb[bestI];                               // LDS broadcast read
            // Cholesky-style L update (done redundantly in all lanes; k <= 4 => registers)
            if (k == 0) {
                Lm[0][0] = 1.f;
            } else {
                float w_[SPARS - 1]; float s2 = 0.f;
                #pragma unroll
                for (int j = 0; j < SPARS - 1; ++j) {
                    if (j < k) {
                        float t = G[I[j] * NUM_EMB + bestI];
                        #pragma unroll
                        for (int p = 0; p < SPARS - 1; ++p)
                            if (p < j) t -= Lm[j][p] * w_[p];
                        w_[j] = t / Lm[j][j];
                        s2 += w_[j] * w_[j];
                    }
                }
                #pragma unroll
                for (int j = 0; j < SPARS - 1; ++j)
                    if (j < k) Lm[k][j] = w_[j];
                Lm[k][k] = sqrtf(fmaxf(1.f - s2, 0.f));
            }
            // forward solve L y = hsel, back solve L^T xs = y  (size k+1)
            float y[SPARS];
            #pragma unroll
            for (int j = 0; j < SPARS; ++j) {
                if (j <= k) {
                    float t = hsel[j];
                    #pragma unroll
                    for (int p = 0; p < SPARS; ++p)
                        if (p < j) t -= Lm[j][p] * y[p];
                    y[j] = t / Lm[j][j];
                }
            }
            #pragma unroll
            for (int j = SPARS - 1; j >= 0; --j) {
                if (j <= k) {
                    float t = y[j];
                    #pragma unroll
                    for (int p = 0; p < SPARS; ++p)
                        if (p > j && p <= k) t -= Lm[p][j] * xs[p];
                    xs[j] = t / Lm[j][j];
                }
            }
        }
        // sparse coefficient scatter (final solve only, matching the reference)
        if (lane == 0) {
            #pragma unroll
            for (int j = 0; j < SPARS; ++j)
                coeff[(size_t)I[j] * BCOLS + b] = xs[j];
        }
        // reconstruction z_dl[:,b] = sum_j xs[j] * Dt[I[j],:]; coalesced via Dt
        #pragma unroll
        for (int half = 0; half < 2; ++half) {
            int m = lane + half * 32;
            float v = 0.f;
            #pragma unroll
            for (int j = 0; j < SPARS; ++j)
                v += xs[j] * Dt[I[j] * EMB_DIM + m];
            int ga = zaddr(m, b);
            float xv = ze[ga];
            zout[ga] = v;                    // z_out == z_dl numerically (straight-through)
            float d = v - xv;
            lacc += d * d;
        }
    }
    #pragma unroll
    for (int off = 16; off > 0; off >>= 1)
        lacc += __shfl_xor(lacc, off);
    if (lane == 0) atomicAdd(acc, lacc);
}

// ---------- kernel 4: finalize loss = (0.25 + 1.0) * MSE ----------
__global__ void dl_loss_kernel(const float* __restrict__ acc, float* __restrict__ lossOut) {
    if (threadIdx.x == 0 && blockIdx.x == 0)
        *lossOut = 1.25f * (*acc) / (float)ZTOTAL;
}

extern "C" void kernel_launch(void* const* d_in, const int* in_sizes, int n_in,
                              void* d_out, int out_size, void* d_ws, size_t ws_size,
                              hipStream_t stream) {
    const float* ze = (const float*)d_in[0];   // (32,64,32,32)
    const float* D  = (const float*)d_in[1];   // (64,512)
    float* zout  = (float*)d_out;              // 2,097,152 floats
    float* lossp = (float*)d_out + ZTOTAL;     // 1 float
    float* coeff = (float*)d_out + ZTOTAL + 1; // 512*32768 floats
    float* G   = (float*)d_ws;                                    // 1 MB
    float* Dt  = (float*)d_ws + NUM_EMB * NUM_EMB;                // 128 KB
    float* acc = (float*)d_ws + NUM_EMB * NUM_EMB + NUM_EMB * EMB_DIM;

    dl_zero_kernel<<<2048, 256, 0, stream>>>(lossp, acc);
    dl_dt_kernel<<<128, 256, 0, stream>>>(D, Dt);
    dl_gram_kernel<<<128, 256, 0, stream>>>(D, G);
    dl_omp_kernel<<<BCOLS / 32, 256, 0, stream>>>(ze, Dt, G, zout, coeff, acc);
    dl_loss_kernel<<<1, 1, 0, stream>>>(acc, lossp);
}